// EGATNodeConv_66383014527134
// MI455X (gfx1250) — compile-verified
//
#include <hip/hip_runtime.h>
#include <hip/hip_bf16.h>

typedef float v2f __attribute__((ext_vector_type(2)));
typedef float v8f __attribute__((ext_vector_type(8)));

#define N_NODES   50000
#define N_EDGES   800000
#define IN_FEATS  256
#define OUT_FEATS 128

// ---------------------------------------------------------------- zero fill
__global__ void gc_zero_kernel(float* __restrict__ p, int n) {
    int i = blockIdx.x * blockDim.x + threadIdx.x;
    int stride = gridDim.x * blockDim.x;
    for (; i < n; i += stride) p[i] = 0.0f;
}

// ---------------------------------------------------------------- degrees
__global__ void gc_degree_kernel(const int* __restrict__ src,
                                 const int* __restrict__ dst,
                                 float* __restrict__ out_deg,
                                 float* __restrict__ in_deg) {
    int e = blockIdx.x * blockDim.x + threadIdx.x;
    if (e >= N_EDGES) return;
    unsafeAtomicAdd(&out_deg[src[e]], 1.0f);
    unsafeAtomicAdd(&in_deg[dst[e]], 1.0f);
}

// ---------------------------------------------------------------- GEMM: h = (x * rsqrt(outdeg)) @ W
// One 256-thread block = 8 waves; block covers 16 rows x 128 cols.
// Wave w computes the 16x16 tile at cols [16w, 16w+16) using V_WMMA_F32_16X16X4_F32.
__global__ void gc_gemm_wmma_kernel(const float* __restrict__ x,
                                    const float* __restrict__ Wm,
                                    const float* __restrict__ out_deg,
                                    float* __restrict__ h) {
    const int lane = threadIdx.x & 31;
    const int wave = threadIdx.x >> 5;      // 0..7 -> N tile
    const int half = lane >> 4;             // 0/1
    const int l    = lane & 15;

    const int m0 = blockIdx.x * 16;         // 3125 blocks * 16 = 50000 exactly
    const int n0 = wave * 16;

    const int row = m0 + l;                                     // A row owned by this lane
    const float rs = rsqrtf(fmaxf(out_deg[row], 1.0f));         // left norm on source feats
    const float* __restrict__ xrow = x + (size_t)row * IN_FEATS;

    v8f c = {};
    #pragma unroll 8
    for (int k0 = 0; k0 < IN_FEATS; k0 += 4) {
        const int ka = k0 + 2 * half;       // this lane's K pair
        v2f a, b;
        a.x = xrow[ka]     * rs;
        a.y = xrow[ka + 1] * rs;
        b.x = Wm[(size_t)ka       * OUT_FEATS + n0 + l];
        b.y = Wm[(size_t)(ka + 1) * OUT_FEATS + n0 + l];
        // (neg_a, A, neg_b, B, c_mod, C, reuse_a, reuse_b)
        c = __builtin_amdgcn_wmma_f32_16x16x4_f32(false, a, false, b,
                                                  (short)0, c, false, false);
    }

    // D layout: VGPR r -> row m0 + r + 8*half, col n0 + l
    #pragma unroll
    for (int r = 0; r < 8; ++r) {
        const int m = m0 + r + 8 * half;
        h[(size_t)m * OUT_FEATS + n0 + l] = c[r];
    }
}

// ---------------------------------------------------------------- SpMM scatter: agg[dst] += h[src] * ew
// One wave (32 lanes) per edge, float4 per lane -> 128 feats.
__global__ void gc_spmm_kernel(const float* __restrict__ h,
                               const float* __restrict__ ew,
                               const int* __restrict__ src,
                               const int* __restrict__ dst,
                               float* __restrict__ agg) {
    const int wave_in_blk = threadIdx.x >> 5;
    const int lane        = threadIdx.x & 31;
    const long long e = (long long)blockIdx.x * (blockDim.x >> 5) + wave_in_blk;
    if (e >= N_EDGES) return;

    const int  s = src[e];
    const int  d = dst[e];
    const float w = ew[e];

    const float4 hv = *(const float4*)(h + (size_t)s * OUT_FEATS + lane * 4);
    float* outp = agg + (size_t)d * OUT_FEATS + lane * 4;
    unsafeAtomicAdd(outp + 0, hv.x * w);
    unsafeAtomicAdd(outp + 1, hv.y * w);
    unsafeAtomicAdd(outp + 2, hv.z * w);
    unsafeAtomicAdd(outp + 3, hv.w * w);
}

// ---------------------------------------------------------------- finalize: out = agg * rsqrt(indeg) + b  (in place)
__global__ void gc_finalize_kernel(float* __restrict__ out,
                                   const float* __restrict__ in_deg,
                                   const float* __restrict__ b) {
    const long long i = (long long)blockIdx.x * blockDim.x + threadIdx.x;
    if (i >= (long long)N_NODES * OUT_FEATS) return;
    const int node = (int)(i >> 7);       // /128
    const int col  = (int)(i & 127);
    const float rn = rsqrtf(fmaxf(in_deg[node], 1.0f));
    out[i] = out[i] * rn + b[col];
}

// ---------------------------------------------------------------- launch
extern "C" void kernel_launch(void* const* d_in, const int* in_sizes, int n_in,
                              void* d_out, int out_size, void* d_ws, size_t ws_size,
                              hipStream_t stream) {
    const float* x   = (const float*)d_in[0];   // [50000, 256]
    const float* ew  = (const float*)d_in[1];   // [800000]
    const float* Wm  = (const float*)d_in[2];   // [256, 128]
    const float* b   = (const float*)d_in[3];   // [128]
    const int*   src = (const int*)d_in[4];     // [800000]
    const int*   dst = (const int*)d_in[5];     // [800000]

    float* out = (float*)d_out;                 // [50000, 128]

    // ws layout: out_deg[50000] | in_deg[50000] | h[50000*128]
    float* out_deg = (float*)d_ws;
    float* in_deg  = out_deg + N_NODES;
    float* h       = in_deg + N_NODES;

    // 1) zero degree buffers and output accumulator
    gc_zero_kernel<<<512, 256, 0, stream>>>(out_deg, 2 * N_NODES);
    gc_zero_kernel<<<4096, 256, 0, stream>>>(out, N_NODES * OUT_FEATS);

    // 2) degree counts
    gc_degree_kernel<<<(N_EDGES + 255) / 256, 256, 0, stream>>>(src, dst, out_deg, in_deg);

    // 3) projection with fp32 WMMA (left-normalized source features)
    gc_gemm_wmma_kernel<<<N_NODES / 16, 256, 0, stream>>>(x, Wm, out_deg, h);

    // 4) edge-weighted gather/scatter-sum (one wave per edge)
    gc_spmm_kernel<<<N_EDGES / 8, 256, 0, stream>>>(h, ew, src, dst, out);

    // 5) right norm + bias, in place on d_out
    gc_finalize_kernel<<<(N_NODES * OUT_FEATS + 255) / 256, 256, 0, stream>>>(out, in_deg, b);
}